// GNNPhysicsODE_90701119357738
// MI455X (gfx1250) — compile-verified
//
#include <hip/hip_runtime.h>
#include <hip/hip_bf16.h>

// ---------------------------------------------------------------------------
// GNNPhysicsODE for MI455X (gfx1250, wave32).
// Dense GEMMs use V_WMMA_F32_16X16X4_F32 (fp32 in / fp32 accum, matches the
// fp32 reference exactly in precision class). Scatter phases use global f32
// atomics. The ODE integrator runs one kernel launch per time step with the
// 2000-zone softmax staged in LDS.
// ---------------------------------------------------------------------------

typedef __attribute__((ext_vector_type(2))) float v2f;
typedef __attribute__((ext_vector_type(8))) float v8f;

static __device__ __forceinline__ float lrelu(float x) { return x > 0.0f ? x : 0.2f * x; }
// order-preserving float<->int transform for integer atomicMax on floats
static __device__ __forceinline__ int f2ord(float f) {
    int i = __float_as_int(f);
    return i ^ ((i >> 31) & 0x7fffffff);
}
static __device__ __forceinline__ float ord2f(int i) {
    int j = i ^ ((i >> 31) & 0x7fffffff);
    return __int_as_float(j);
}

// ---------------------------------------------------------------------------
// out[n, Nout] = act(X[n, K] @ W[Nout, K]^T + bias)
// One wave computes one 16x16 output tile via WMMA_F32_16X16X4_F32.
// grid = (n/16, Nout/64), block = 128 (4 waves -> 64 output columns / block).
// flags: bit0 = add bias, bit1 = relu. Requires n % 16 == 0, K % 4 == 0.
// ---------------------------------------------------------------------------
__global__ __launch_bounds__(128) void gemm_wmma_kernel(
    const float* __restrict__ X, const float* __restrict__ W,
    const float* __restrict__ bias, float* __restrict__ out,
    int K, int Nout, int flags)
{
    int wave = threadIdx.x >> 5;
    int lane = threadIdx.x & 31;
    int row0 = blockIdx.x * 16;
    int col0 = blockIdx.y * 64 + wave * 16;
    int hi   = lane >> 4;      // which half of the wave
    int r    = lane & 15;

    // A fragment: lanes 0-15 hold K={0,1}, lanes 16-31 hold K={2,3} per 4-step.
    // B (K x 16) mirrors A because both X and W are indexed [row][k].
    const float* xrow = X + (size_t)(row0 + r) * K + 2 * hi;
    const float* wrow = W + (size_t)(col0 + r) * K + 2 * hi;

    v8f c = {0.f, 0.f, 0.f, 0.f, 0.f, 0.f, 0.f, 0.f};
    for (int k0 = 0; k0 < K; k0 += 4) {
        v2f a = *(const v2f*)(xrow + k0);
        v2f b = *(const v2f*)(wrow + k0);
        c = __builtin_amdgcn_wmma_f32_16x16x4_f32(false, a, false, b, (short)0, c,
                                                  false, false);
    }

    float bv = (flags & 1) ? bias[col0 + r] : 0.0f;
#pragma unroll
    for (int i = 0; i < 8; ++i) {
        // D layout: VGPR i -> M = i (lanes 0-15) / M = 8+i (lanes 16-31), N = lane%16
        int row = row0 + 8 * hi + i;
        float v = c[i] + bv;
        if (flags & 2) v = fmaxf(v, 0.0f);
        out[(size_t)row * Nout + (col0 + r)] = v;
    }
}

// ------------------------------ small utility kernels ----------------------
__global__ void fill_kernel(float* p, float v, int n) {
    int i = blockIdx.x * 256 + threadIdx.x;
    if (i < n) p[i] = v;
}
__global__ void deg_kernel(const int* __restrict__ dst, float* deg, int E) {
    int i = blockIdx.x * 256 + threadIdx.x;
    if (i < E) atomicAdd(&deg[dst[i]], 1.0f);
}
__global__ void rsqrt_kernel(const float* deg, float* dinv, int n) {
    int i = blockIdx.x * 256 + threadIdx.x;
    if (i < n) dinv[i] = rsqrtf(fmaxf(deg[i], 1e-12f));
}
__global__ void relu_kernel(float* x, int n) {
    int i = blockIdx.x * 256 + threadIdx.x;
    if (i < n) x[i] = fmaxf(x[i], 0.0f);
}
// out = hw * dinv^2 (self loop) + bias   (total = n*64)
__global__ void gcn_init_kernel(const float* __restrict__ hw, const float* __restrict__ dinv,
                                const float* __restrict__ b, float* __restrict__ out, int total) {
    int i = blockIdx.x * 256 + threadIdx.x;
    if (i < total) {
        int r = i >> 6, c = i & 63;
        float dv = dinv[r];
        out[i] = hw[i] * dv * dv + b[c];
    }
}
// grid = E, block = 64: out[d] += hw[s] * dinv[s]*dinv[d]
__global__ void gcn_scatter_kernel(const int* __restrict__ src, const int* __restrict__ dst,
                                   const float* __restrict__ hw, const float* __restrict__ dinv,
                                   float* __restrict__ out) {
    int e = blockIdx.x, c = threadIdx.x;
    int s = src[e], d = dst[e];
    float nm = dinv[s] * dinv[d];
    atomicAdd(&out[(size_t)d * 64 + c], hw[(size_t)s * 64 + c] * nm);
}

// ------------------------------ GAT kernels --------------------------------
__global__ void gat_attn_kernel(const float* __restrict__ X4, const float* __restrict__ atts,
                                const float* __restrict__ attd, float* AS, float* AD, int total) {
    int i = blockIdx.x * 256 + threadIdx.x;
    if (i < total) {
        int node = i >> 2, h = i & 3;
        const float* xv = X4 + (size_t)node * 256 + h * 64;
        const float* s = atts + h * 64;
        const float* d = attd + h * 64;
        float a = 0.f, b = 0.f;
        for (int k = 0; k < 64; ++k) { float v = xv[k]; a += v * s[k]; b += v * d[k]; }
        AS[i] = a; AD[i] = b;
    }
}
__global__ void gat_minit_kernel(const float* AS, const float* AD, int* M, int total) {
    int i = blockIdx.x * 256 + threadIdx.x;
    if (i < total) M[i] = f2ord(lrelu(AS[i] + AD[i]));   // self-loop alpha
}
__global__ void gat_emax_kernel(const int* src, const int* dst, const float* AS,
                                const float* AD, int* M, int total /* E*4 */) {
    int i = blockIdx.x * 256 + threadIdx.x;
    if (i < total) {
        int e = i >> 2, h = i & 3;
        int s = src[e], d = dst[e];
        float v = lrelu(AS[s * 4 + h] + AD[d * 4 + h]);
        atomicMax(&M[d * 4 + h], f2ord(v));
    }
}
__global__ void gat_zinit_kernel(const float* AS, const float* AD, const int* M,
                                 float* Z, int total) {
    int i = blockIdx.x * 256 + threadIdx.x;
    if (i < total) Z[i] = expf(lrelu(AS[i] + AD[i]) - ord2f(M[i]));
}
__global__ void gat_zscatter_kernel(const int* src, const int* dst, const float* AS,
                                    const float* AD, const int* M, float* Z, int total) {
    int i = blockIdx.x * 256 + threadIdx.x;
    if (i < total) {
        int e = i >> 2, h = i & 3;
        int s = src[e], d = dst[e];
        int a = d * 4 + h;
        float v = lrelu(AS[s * 4 + h] + AD[a]);
        atomicAdd(&Z[a], expf(v - ord2f(M[a])));
    }
}
__global__ void gat_oinit_kernel(const float* X4, const float* AS, const float* AD,
                                 const int* M, const float* Z, float* O4, int total) {
    int i = blockIdx.x * 256 + threadIdx.x;
    if (i < total) {
        int node = i >> 8, h = (i >> 6) & 3;
        int a = node * 4 + h;
        float w = expf(lrelu(AS[a] + AD[a]) - ord2f(M[a])) / (Z[a] + 1e-16f);
        O4[i] = X4[i] * w;
    }
}
// grid = E, block = 256 (4 heads x 64 dims)
__global__ void gat_oscatter_kernel(const int* __restrict__ src, const int* __restrict__ dst,
                                    const float* __restrict__ X4, const float* __restrict__ AS,
                                    const float* __restrict__ AD, const int* __restrict__ M,
                                    const float* __restrict__ Z, float* __restrict__ O4) {
    int e = blockIdx.x, t = threadIdx.x;
    int h = t >> 6;
    int s = src[e], d = dst[e];
    int a = d * 4 + h;
    float v = lrelu(AS[s * 4 + h] + AD[a]);
    float w = expf(v - ord2f(M[a])) / (Z[a] + 1e-16f);
    atomicAdd(&O4[(size_t)d * 256 + t], X4[(size_t)s * 256 + t] * w);
}
__global__ void gat_mean_kernel(const float* O4, const float* gb, float* out, int total) {
    int i = blockIdx.x * 256 + threadIdx.x;
    if (i < total) {
        int node = i >> 6, c = i & 63;
        const float* o = O4 + (size_t)node * 256;
        out[i] = 0.25f * (o[c] + o[64 + c] + o[128 + c] + o[192 + c]) + gb[c];
    }
}

// ------------------------------ misc precompute ----------------------------
__global__ void locsq_kernel(const float* loce, float* lsq, int n) {
    int i = blockIdx.x * 256 + threadIdx.x;
    if (i < n) {
        const float* p = loce + (size_t)i * 64;
        float s = 0.f;
        for (int k = 0; k < 64; ++k) s += p[k] * p[k];
        lsq[i] = s;
    }
}
__global__ void adj_init_kernel(unsigned char* adj, int NL, int total) {
    int i = blockIdx.x * 256 + threadIdx.x;
    if (i < total) adj[i] = (i % (NL + 1) == 0) ? 1 : 0;   // identity diagonal
}
__global__ void adj_set_kernel(const int* src, const int* dst, unsigned char* adj,
                               int NL, int E) {
    int i = blockIdx.x * 256 + threadIdx.x;
    if (i < E) adj[(size_t)src[i] * NL + dst[i]] = 1;
}
__global__ void gather_kernel(const float* table, const int* idx, float* out0,
                              float* out1, int total) {
    int i = blockIdx.x * 256 + threadIdx.x;
    if (i < total) {
        int r = i >> 6, c = i & 63;
        float v = table[(size_t)idx[r] * 64 + c];
        out0[i] = v;
        if (out1) out1[i] = v;
    }
}

// ---------------------------------------------------------------------------
// One ODE step. grid = B blocks (one per batch row), block = 256.
// Reads times[] on device (no host readback -> graph-capture safe).
// ---------------------------------------------------------------------------
__global__ __launch_bounds__(256) void ode_step_kernel(
    float* __restrict__ state, const float* __restrict__ pemb,
    const float* __restrict__ loce, const float* __restrict__ lsq,
    const unsigned char* __restrict__ adj,
    const float* __restrict__ t1w, const float* __restrict__ t1b,
    const float* __restrict__ t2w, const float* __restrict__ t2b,
    const float* __restrict__ t3w, const float* __restrict__ t3b,
    const float* __restrict__ f1w, const float* __restrict__ f1b,
    const float* __restrict__ f2w, const float* __restrict__ f2b,
    const float* __restrict__ f3w, const float* __restrict__ f3b,
    const float* __restrict__ times, int step, int NL,
    float* __restrict__ traj_out)
{
    __shared__ float s_inp[192];            // [state | pemb | tenc]
    __shared__ float s_t1[32], s_t2[32];
    __shared__ float s_h1[128], s_h2[64], s_v[64], s_cc[64];
    __shared__ float s_logit[2048], s_invn[2048];
    __shared__ float s_rf[256];
    __shared__ int   s_ri[256];
    __shared__ float s_max, s_sum;
    __shared__ int   s_cur, s_cnt;

    int b = blockIdx.x, tid = threadIdx.x;
    float t  = times[step];
    float dt = times[step] - times[step - 1];

    if (tid < 64)       s_inp[tid] = state[b * 64 + tid];
    else if (tid < 128) s_inp[tid] = pemb[b * 64 + (tid - 64)];
    if (tid < 32) s_t1[tid] = fmaxf(t1w[tid] * t + t1b[tid], 0.f);
    __syncthreads();
    if (tid < 32) {
        float a = t2b[tid];
        for (int k = 0; k < 32; ++k) a += t2w[tid * 32 + k] * s_t1[k];
        s_t2[tid] = fmaxf(a, 0.f);
    }
    __syncthreads();
    if (tid < 64) {
        float a = t3b[tid];
        for (int k = 0; k < 32; ++k) a += t3w[tid * 32 + k] * s_t2[k];
        s_inp[128 + tid] = a;
    }
    __syncthreads();
    // flow MLP: 192 -> 128 -> 64 -> 64
    if (tid < 128) {
        float a = f1b[tid];
        for (int k = 0; k < 192; ++k) a += f1w[tid * 192 + k] * s_inp[k];
        s_h1[tid] = fmaxf(a, 0.f);
    }
    __syncthreads();
    if (tid < 64) {
        float a = f2b[tid];
        for (int k = 0; k < 128; ++k) a += f2w[tid * 128 + k] * s_h1[k];
        s_h2[tid] = fmaxf(a, 0.f);
    }
    __syncthreads();
    if (tid < 64) {
        float a = f3b[tid];
        for (int k = 0; k < 64; ++k) a += f3w[tid * 64 + k] * s_h2[k];
        s_v[tid] = a;
    }
    __syncthreads();

    // nearest zone: argmin_j (lsq[j] - 2 * <state, loc_j>)  (first-index ties)
    float best = 3.4e38f; int bj = 0;
    for (int j = tid; j < NL; j += 256) {
        const float* le = loce + (size_t)j * 64;
        float dot = 0.f;
        for (int k = 0; k < 64; ++k) dot += s_inp[k] * le[k];
        float sc = lsq[j] - 2.f * dot;
        if (sc < best) { best = sc; bj = j; }
    }
    s_rf[tid] = best; s_ri[tid] = bj;
    __syncthreads();
    for (int s = 128; s > 0; s >>= 1) {
        if (tid < s) {
            float ov = s_rf[tid + s]; int oi = s_ri[tid + s];
            if (ov < s_rf[tid] || (ov == s_rf[tid] && oi < s_ri[tid])) {
                s_rf[tid] = ov; s_ri[tid] = oi;
            }
        }
        __syncthreads();
    }
    if (tid == 0) s_cur = s_ri[0];
    __syncthreads();
    int cur = s_cur;
    if (tid < 64) s_cc[tid] = loce[(size_t)cur * 64 + tid];
    __syncthreads();

    // masked logits + inverse norms, staged in LDS
    float lmax = -3.4e38f; int cnt = 0;
    const unsigned char* arow = adj + (size_t)cur * NL;
    for (int j = tid; j < NL; j += 256) {
        const float* le = loce + (size_t)j * 64;
        float n2 = 0.f, pd = 0.f;
        for (int k = 0; k < 64; ++k) {
            float dk = le[k] - s_cc[k];
            n2 += dk * dk;
            pd += dk * s_v[k];
        }
        float nrm  = sqrtf(n2);
        float invn = (nrm > 1e-6f) ? 1.0f / fmaxf(nrm, 1e-12f) : 0.0f;
        bool  m    = arow[j] != 0;
        float lg   = m ? pd * invn : -1e30f;
        s_logit[j] = lg; s_invn[j] = invn;
        if (m) cnt++;
        lmax = fmaxf(lmax, lg);
    }
    s_rf[tid] = lmax; s_ri[tid] = cnt;
    __syncthreads();
    for (int s = 128; s > 0; s >>= 1) {
        if (tid < s) { s_rf[tid] = fmaxf(s_rf[tid], s_rf[tid + s]); s_ri[tid] += s_ri[tid + s]; }
        __syncthreads();
    }
    if (tid == 0) { s_max = s_rf[0]; s_cnt = s_ri[0]; }
    __syncthreads();
    float mx = s_max;
    float lsum = 0.f;
    for (int j = tid; j < NL; j += 256) lsum += expf(s_logit[j] - mx);
    s_rf[tid] = lsum;
    __syncthreads();
    for (int s = 128; s > 0; s >>= 1) {
        if (tid < s) s_rf[tid] += s_rf[tid + s];
        __syncthreads();
    }
    if (tid == 0) s_sum = s_rf[0];
    __syncthreads();

    // cv_d = sum_j softmax_j * dir_{j,d}; thread tid<64 owns dim d (coalesced loce reads)
    if (tid < 64) {
        float cv = 0.f;
        float cc = s_cc[tid];
        if (s_cnt > 1) {
            float sum = s_sum;
            for (int j = 0; j < NL; ++j) {
                float w = expf(s_logit[j] - mx);
                cv += w * (loce[(size_t)j * 64 + tid] - cc) * s_invn[j];
            }
            cv /= sum;
        }
        float ns = s_inp[tid] + cv * 0.5f * dt;
        state[b * 64 + tid]    = ns;
        traj_out[b * 64 + tid] = ns;
    }
}

// zones[tb] = argmin_j (lsq[j] - 2 * <traj[tb], loc_j>), grid = T*B, block = 256
__global__ __launch_bounds__(256) void zones_kernel(
    const float* __restrict__ traj, const float* __restrict__ loce,
    const float* __restrict__ lsq, float* __restrict__ zout, int NL)
{
    __shared__ float s_st[64];
    __shared__ float s_rf[256];
    __shared__ int   s_ri[256];
    int tb = blockIdx.x, tid = threadIdx.x;
    if (tid < 64) s_st[tid] = traj[(size_t)tb * 64 + tid];
    __syncthreads();
    float best = 3.4e38f; int bj = 0;
    for (int j = tid; j < NL; j += 256) {
        const float* le = loce + (size_t)j * 64;
        float dot = 0.f;
        for (int k = 0; k < 64; ++k) dot += s_st[k] * le[k];
        float sc = lsq[j] - 2.f * dot;
        if (sc < best) { best = sc; bj = j; }
    }
    s_rf[tid] = best; s_ri[tid] = bj;
    __syncthreads();
    for (int s = 128; s > 0; s >>= 1) {
        if (tid < s) {
            float ov = s_rf[tid + s]; int oi = s_ri[tid + s];
            if (ov < s_rf[tid] || (ov == s_rf[tid] && oi < s_ri[tid])) {
                s_rf[tid] = ov; s_ri[tid] = oi;
            }
        }
        __syncthreads();
    }
    if (tid == 0) zout[tb] = (float)s_ri[0];
}

// ---------------------------------------------------------------------------
static inline int cdiv(int a, int b) { return (a + b - 1) / b; }

extern "C" void kernel_launch(void* const* d_in, const int* in_sizes, int n_in,
                              void* d_out, int out_size, void* d_ws, size_t ws_size,
                              hipStream_t stream)
{
    const float* x_loc = (const float*)d_in[0];
    const int*   ei_l  = (const int*)d_in[1];
    const float* x_per = (const float*)d_in[2];
    const int*   ei_p  = (const int*)d_in[3];
    const float* times = (const float*)d_in[4];
    const int*   pid   = (const int*)d_in[5];
    const int*   iz    = (const int*)d_in[6];

    const int NL = in_sizes[0] / 16;   // 2000
    const int EL = in_sizes[1] / 2;    // 16000
    const int NP = in_sizes[2] / 32;   // 50000
    const int EP = in_sizes[3] / 2;    // 400000
    const int T  = in_sizes[4];        // 24
    const int B  = in_sizes[5];        // 64

    // params flattened in dict-insertion order:
    // loc[12]: in.w,in.b,gcn1.w,gcn1.b,gcn2.w,gcn2.b,gat_w,gat_b,att_src,att_dst,out.w,out.b
    // per[12]: same; time[6]: l1.w,l1.b,l2.w,l2.b,l3.w,l3.b; flow[6]: same
    const float* P[36];
    for (int i = 0; i < 36; ++i) P[i] = (const float*)d_in[7 + i];

    // ---- workspace bump allocator (sized for the person graph, reused for loc)
    char* ws = (char*)d_ws;
    size_t off = 0;
    auto alloc = [&](size_t bytes) -> char* {
        char* p = ws + off;
        off = (off + bytes + 255) & ~(size_t)255;
        return p;
    };
    float* H    = (float*)alloc((size_t)NP * 64 * 4);
    float* HW   = (float*)alloc((size_t)NP * 64 * 4);
    float* H2   = (float*)alloc((size_t)NP * 64 * 4);
    float* X4   = (float*)alloc((size_t)NP * 256 * 4);
    float* O4   = (float*)alloc((size_t)NP * 256 * 4);
    float* AS   = (float*)alloc((size_t)NP * 4 * 4);
    float* AD   = (float*)alloc((size_t)NP * 4 * 4);
    int*   MT   = (int*)  alloc((size_t)NP * 4 * 4);
    float* ZZ   = (float*)alloc((size_t)NP * 4 * 4);
    float* DEG  = (float*)alloc((size_t)NP * 4);
    float* DINV = (float*)alloc((size_t)NP * 4);
    float* LOCE = (float*)alloc((size_t)NL * 64 * 4);
    float* LSQ  = (float*)alloc((size_t)NL * 4);
    unsigned char* ADJ = (unsigned char*)alloc((size_t)NL * NL);
    float* PEMB  = (float*)alloc((size_t)B * 64 * 4);
    float* STATE = (float*)alloc((size_t)B * 64 * 4);
    (void)ws_size; (void)n_in; (void)out_size;

    // ---- full GNN embed pipeline (shared scratch); result in emb_out
    auto embed = [&](const float* x, int fin, int n, int E,
                     const int* src, const int* dst,
                     const float* const* pp, float* emb_out) {
        // input linear + relu
        gemm_wmma_kernel<<<dim3(n / 16, 1), 128, 0, stream>>>(x, pp[0], pp[1], H, fin, 64, 3);
        // GCN normalization (deg includes the self loop)
        fill_kernel<<<cdiv(n, 256), 256, 0, stream>>>(DEG, 1.0f, n);
        deg_kernel<<<cdiv(E, 256), 256, 0, stream>>>(dst, DEG, E);
        rsqrt_kernel<<<cdiv(n, 256), 256, 0, stream>>>(DEG, DINV, n);
        // GCN layer 1: H -> H2
        gemm_wmma_kernel<<<dim3(n / 16, 1), 128, 0, stream>>>(H, pp[2], nullptr, HW, 64, 64, 0);
        gcn_init_kernel<<<cdiv(n * 64, 256), 256, 0, stream>>>(HW, DINV, pp[3], H2, n * 64);
        gcn_scatter_kernel<<<E, 64, 0, stream>>>(src, dst, HW, DINV, H2);
        relu_kernel<<<cdiv(n * 64, 256), 256, 0, stream>>>(H2, n * 64);
        // GCN layer 2: H2 -> H
        gemm_wmma_kernel<<<dim3(n / 16, 1), 128, 0, stream>>>(H2, pp[4], nullptr, HW, 64, 64, 0);
        gcn_init_kernel<<<cdiv(n * 64, 256), 256, 0, stream>>>(HW, DINV, pp[5], H, n * 64);
        gcn_scatter_kernel<<<E, 64, 0, stream>>>(src, dst, HW, DINV, H);
        relu_kernel<<<cdiv(n * 64, 256), 256, 0, stream>>>(H, n * 64);
        // GAT: x = H @ gat_w^T -> [n, 4, 64]
        gemm_wmma_kernel<<<dim3(n / 16, 4), 128, 0, stream>>>(H, pp[6], nullptr, X4, 64, 256, 0);
        gat_attn_kernel<<<cdiv(n * 4, 256), 256, 0, stream>>>(X4, pp[8], pp[9], AS, AD, n * 4);
        gat_minit_kernel<<<cdiv(n * 4, 256), 256, 0, stream>>>(AS, AD, MT, n * 4);
        gat_emax_kernel<<<cdiv(E * 4, 256), 256, 0, stream>>>(src, dst, AS, AD, MT, E * 4);
        gat_zinit_kernel<<<cdiv(n * 4, 256), 256, 0, stream>>>(AS, AD, MT, ZZ, n * 4);
        gat_zscatter_kernel<<<cdiv(E * 4, 256), 256, 0, stream>>>(src, dst, AS, AD, MT, ZZ, E * 4);
        gat_oinit_kernel<<<cdiv(n * 256, 256), 256, 0, stream>>>(X4, AS, AD, MT, ZZ, O4, n * 256);
        gat_oscatter_kernel<<<E, 256, 0, stream>>>(src, dst, X4, AS, AD, MT, ZZ, O4);
        gat_mean_kernel<<<cdiv(n * 64, 256), 256, 0, stream>>>(O4, pp[7], H2, n * 64);
        // output linear (+bias, no relu)
        gemm_wmma_kernel<<<dim3(n / 16, 1), 128, 0, stream>>>(H2, pp[10], pp[11], emb_out, 64, 64, 1);
    };

    // person embeddings (result left in HW, free at that point), gather pemb
    embed(x_per, 32, NP, EP, ei_p, ei_p + EP, &P[12], HW);
    gather_kernel<<<cdiv(B * 64, 256), 256, 0, stream>>>(HW, pid, PEMB, nullptr, B * 64);

    // location embeddings + precompute
    embed(x_loc, 16, NL, EL, ei_l, ei_l + EL, &P[0], LOCE);
    locsq_kernel<<<cdiv(NL, 256), 256, 0, stream>>>(LOCE, LSQ, NL);
    adj_init_kernel<<<cdiv(NL * NL, 256), 256, 0, stream>>>(ADJ, NL, NL * NL);
    adj_set_kernel<<<cdiv(EL, 256), 256, 0, stream>>>(ei_l, ei_l + EL, ADJ, NL, EL);

    // initial state = loc_emb[initial_zone]; also traj[0]
    float* traj = (float*)d_out;
    gather_kernel<<<cdiv(B * 64, 256), 256, 0, stream>>>(LOCE, iz, STATE, traj, B * 64);

    // ODE integration (one launch per step; states stream into d_out)
    for (int s = 1; s < T; ++s) {
        ode_step_kernel<<<B, 256, 0, stream>>>(
            STATE, PEMB, LOCE, LSQ, ADJ,
            P[24], P[25], P[26], P[27], P[28], P[29],
            P[30], P[31], P[32], P[33], P[34], P[35],
            times, s, NL, traj + (size_t)s * B * 64);
    }

    // zones over the whole trajectory, appended after traj in d_out
    zones_kernel<<<T * B, 256, 0, stream>>>(traj, LOCE, LSQ,
                                            traj + (size_t)T * B * 64, NL);
}